// Attention_17128329576773
// MI455X (gfx1250) — compile-verified
//
#include <hip/hip_runtime.h>

#define B_    32
#define H_    32
#define KVH_  8
#define D_    128
#define BS_   16
#define BPS_  256           // blocks per sequence
#define NWAVES 8
#define SCALE_ 0.08838834764831845f

typedef float v2f __attribute__((ext_vector_type(2)));
typedef float v8f __attribute__((ext_vector_type(8)));
typedef unsigned int v4u __attribute__((ext_vector_type(4)));
typedef int v4i __attribute__((ext_vector_type(4)));
typedef int v8i __attribute__((ext_vector_type(8)));

// SWAPX16: exchange value with lane^16 (group-of-32 mode, xor=0x10, and=0x1f)
__device__ __forceinline__ float swz16(float x) {
  return __int_as_float(__builtin_amdgcn_ds_swizzle(__float_as_int(x), 0x401F));
}

// TDM: async-load a 16x128 fp32 tile (row stride KVH_*D_ elems) into LDS.
// D# Group0: count=1 | lds_addr | global_addr[56:0] | type=2
// D# Group1: data_size=4B, tensor_dim0=128, tensor_dim1=16, tile=128x16,
//            tensor_dim0_stride=1024. Groups 2/3 (and trailing group) zero: 2D tile.
__device__ __forceinline__ void tdm_load_tile(const float* gsrc, float* ldst) {
  const unsigned long long ga = (unsigned long long)(uintptr_t)gsrc;
  unsigned int lo = (unsigned int)(uintptr_t)ldst;          // LDS byte offset
  lo = (unsigned int)__builtin_amdgcn_readfirstlane((int)lo);
  v4u g0;
  g0.x = 1u;                                                // count=1, user D#
  g0.y = lo;                                                // lds_addr
  g0.z = (unsigned int)(ga & 0xffffffffull);                // global_addr[31:0]
  g0.w = (unsigned int)((ga >> 32) & 0x1ffffffull) | 0x80000000u; // [56:32]|type=2
  v8i g1;
  g1[0] = 0x00020000;            // data_size=2 (4 bytes); no multicast/flags
  g1[1] = (int)(128u << 16);     // tensor_dim0[15:0]=128
  g1[2] = (int)(16u << 16);      // tensor_dim0 hi=0 | tensor_dim1[15:0]=16
  g1[3] = (int)(128u << 16);     // tensor_dim1 hi=0 | tile_dim0=128
  g1[4] = 16;                    // tile_dim1=16, tile_dim2=0
  g1[5] = (int)(KVH_ * D_);      // tensor_dim0_stride=1024 (elements)
  g1[6] = 0;                     // stride hi | tensor_dim1_stride lo
  g1[7] = 0;
  v4i gz4 = (v4i){0, 0, 0, 0};
  v8i gz8 = (v8i){0, 0, 0, 0, 0, 0, 0, 0};
  __builtin_amdgcn_tensor_load_to_lds(g0, g1, gz4, gz4, gz8, 0);
}

__global__ __launch_bounds__(256, 1)
void paged_attn_wmma_kernel(const float* __restrict__ q,
                            const float* __restrict__ knew,
                            const float* __restrict__ vnew,
                            const float* __restrict__ kcache,
                            const float* __restrict__ vcache,
                            const int*   __restrict__ btab,
                            const int*   __restrict__ slens,
                            float*       __restrict__ out)
{
  extern __shared__ float smem[];
  const int tid  = threadIdx.x;
  const int wave = tid >> 5;
  const int lane = tid & 31;
  const int lm   = lane & 15;     // lane mod 16
  const int hi   = lane >> 4;     // 0 = low half, 1 = high half

  const int wg  = blockIdx.x;
  const int b   = wg / KVH_;
  const int kvh = wg % KVH_;

  // LDS layout (floats):
  float* ldsKV  = smem;                     // 8 waves * 2 bufs * 4096 (K|V)
  float* ldsQ   = smem + NWAVES * 8192;     // 16 heads * 128 (heads 4..15 zero)
  float* ldsAcc = ldsQ + 16 * D_;           // 8 waves * 4 heads * 128
  float* ldsM   = ldsAcc + NWAVES * 4 * D_; // 32
  float* ldsL   = ldsM + 32;                // 32

  // ---- stage Q (4 real heads of this kv group, zero-pad to 16 rows) ----
  for (int i = tid; i < 16 * D_; i += 256) {
    const int h16 = i >> 7;
    const int d   = i & (D_ - 1);
    float v = 0.0f;
    if (h16 < 4) v = q[((size_t)b * H_ + (kvh * 4 + h16)) * D_ + d];
    ldsQ[i] = v;
  }
  __syncthreads();

  int seq = slens[b]; if (seq < 1) seq = 1;
  const int pos    = seq - 1;
  const int posblk = pos >> 4;
  const int posrow = pos & 15;
  const int nblk   = (seq + BS_ - 1) >> 4;

  float* wbase = ldsKV + wave * 8192;   // 2 buffers of 4096 floats (K 2048 | V 2048)

  float mrun = -3.0e38f;
  float lrun = 0.0f;
  v8f acc[8];
  #pragma unroll
  for (int t = 0; t < 8; ++t) acc[t] = v8f{};

  // prologue: async-load this wave's first block pair via TDM
  if (wave < nblk) {
    const int p0 = __builtin_amdgcn_readfirstlane(btab[b * BPS_ + wave]);
    tdm_load_tile(kcache + ((size_t)p0 * BS_ * KVH_ + kvh) * D_, wbase);
    tdm_load_tile(vcache + ((size_t)p0 * BS_ * KVH_ + kvh) * D_, wbase + 2048);
  }

  int bufsel = 0;
  for (int sblk = wave; sblk < nblk; sblk += NWAVES) {
    float* ldsK = wbase + bufsel * 4096;
    float* ldsV = ldsK + 2048;

    // issue next block's TDM pair into the other buffer, then wait for current
    const int nxt = sblk + NWAVES;
    if (nxt < nblk) {
      const int pn = __builtin_amdgcn_readfirstlane(btab[b * BPS_ + nxt]);
      float* nb = wbase + (bufsel ^ 1) * 4096;
      tdm_load_tile(kcache + ((size_t)pn * BS_ * KVH_ + kvh) * D_, nb);
      tdm_load_tile(vcache + ((size_t)pn * BS_ * KVH_ + kvh) * D_, nb + 2048);
      __builtin_amdgcn_s_wait_tensorcnt(2);   // current pair landed (in-order)
    } else {
      __builtin_amdgcn_s_wait_tensorcnt(0);
    }
    asm volatile("" ::: "memory");            // keep ds reads below the wait

    // splice the new token's K/V row (reference writes it into the cache)
    if (sblk == posblk) {
      float4 kk = *(const float4*)(knew + ((size_t)b * KVH_ + kvh) * D_ + lane * 4);
      float4 vv = *(const float4*)(vnew + ((size_t)b * KVH_ + kvh) * D_ + lane * 4);
      *(float4*)(ldsK + posrow * D_ + lane * 4) = kk;
      *(float4*)(ldsV + posrow * D_ + lane * 4) = vv;
    }

    // ---- scores[16 tok x 16 head] = K_tile(16x128) @ Q^T(128x16) ----
    // A: lane lm = token row, regs = d {4c,4c+1} (lo) / {4c+2,4c+3} (hi)
    // B: lane lm = head col,  regs = d likewise (from zero-padded ldsQ)
    v8f sc = v8f{};
    const float* aBase = ldsK + lm * D_ + hi * 2;
    const float* bBase = ldsQ + lm * D_ + hi * 2;
    #pragma unroll
    for (int c = 0; c < 32; ++c) {
      v2f a  = *(const v2f*)(aBase + c * 4);
      v2f bq = *(const v2f*)(bBase + c * 4);
      sc = __builtin_amdgcn_wmma_f32_16x16x4_f32(false, a, false, bq,
                                                 (short)0, sc, false, false);
    }

    // ---- online softmax (row = head on lane; tokens split lane pair) ----
    float pr[8];
    float m8 = -3.0e38f;
    const int tokbase = sblk * BS_ + hi * 8;
    #pragma unroll
    for (int r = 0; r < 8; ++r) {
      float v = sc[r] * SCALE_;
      v = (tokbase + r < seq) ? v : -3.0e38f;
      pr[r] = v;
      m8 = fmaxf(m8, v);
    }
    const float mblk = fmaxf(m8, swz16(m8));      // max over all 16 tokens
    const float mnew = fmaxf(mrun, mblk);
    const float alpha = __expf(mrun - mnew);
    float lsum = 0.0f;
    #pragma unroll
    for (int r = 0; r < 8; ++r) {
      pr[r] = __expf(pr[r] - mnew);
      lsum += pr[r];
    }
    lrun = lrun * alpha + lsum + swz16(lsum);
    mrun = mnew;
    #pragma unroll
    for (int t = 0; t < 8; ++t)
      #pragma unroll
      for (int r = 0; r < 8; ++r) acc[t][r] *= alpha;   // per-lane head scale

    // ---- reshape probs (score layout) -> B fragments probs^T[4 keys x 16 heads]
    float z0 = swz16(pr[0]), z1 = swz16(pr[1]), z2 = swz16(pr[2]), z3 = swz16(pr[3]);
    float z4 = swz16(pr[4]), z5 = swz16(pr[5]), z6 = swz16(pr[6]), z7 = swz16(pr[7]);
    v2f bf[4];
    bf[0].x = hi ? z2    : pr[0];  bf[0].y = hi ? z3    : pr[1];  // keys 0..3
    bf[1].x = hi ? z6    : pr[4];  bf[1].y = hi ? z7    : pr[5];  // keys 4..7
    bf[2].x = hi ? pr[2] : z0;     bf[2].y = hi ? pr[3] : z1;     // keys 8..11
    bf[3].x = hi ? pr[6] : z4;     bf[3].y = hi ? pr[7] : z5;     // keys 12..15

    // ---- out^T[16 d x 16 head] += V^T(16x4) @ probs^T(4x16), 8 d-tiles ----
    #pragma unroll
    for (int t = 0; t < 8; ++t) {
      const float* vB = ldsV + hi * 2 * D_ + t * 16 + lm;
      #pragma unroll
      for (int c = 0; c < 4; ++c) {
        v2f a;
        a.x = vB[(4 * c)     * D_];   // V[key 4c+2*hi,   d = 16t+lm]
        a.y = vB[(4 * c + 1) * D_];   // V[key 4c+2*hi+1, d = 16t+lm]
        acc[t] = __builtin_amdgcn_wmma_f32_16x16x4_f32(false, a, false, bf[c],
                                                       (short)0, acc[t], false, false);
      }
    }

    bufsel ^= 1;
  }

  // ---- dump per-wave partials (only head lanes 0..3 / 16..19 are real) ----
  if (lm < 4) {
    const int hh = wave * 4 + lm;
    if (!hi) { ldsM[hh] = mrun; ldsL[hh] = lrun; }
    #pragma unroll
    for (int t = 0; t < 8; ++t)
      #pragma unroll
      for (int r = 0; r < 8; ++r)
        ldsAcc[hh * D_ + t * 16 + r + hi * 8] = acc[t][r];
  }
  __syncthreads();

  // ---- merge 8 wave partials (flash combine) and write output ----
  for (int pidx = tid; pidx < 4 * D_; pidx += 256) {
    const int h = pidx >> 7;
    const int d = pidx & (D_ - 1);
    float M = -3.0e38f;
    #pragma unroll
    for (int w = 0; w < NWAVES; ++w) M = fmaxf(M, ldsM[w * 4 + h]);
    float den = 0.0f, num = 0.0f;
    #pragma unroll
    for (int w = 0; w < NWAVES; ++w) {
      const float e = __expf(ldsM[w * 4 + h] - M);
      den += ldsL[w * 4 + h] * e;
      num += ldsAcc[(w * 4 + h) * D_ + d] * e;
    }
    out[((size_t)b * H_ + kvh * 4 + h) * D_ + d] = num / den;
  }
}

extern "C" void kernel_launch(void* const* d_in, const int* in_sizes, int n_in,
                              void* d_out, int out_size, void* d_ws, size_t ws_size,
                              hipStream_t stream) {
  const float* q  = (const float*)d_in[0];
  const float* kn = (const float*)d_in[1];
  const float* vn = (const float*)d_in[2];
  const float* kc = (const float*)d_in[3];
  const float* vc = (const float*)d_in[4];
  const int*   bt = (const int*)d_in[5];
  const int*   sl = (const int*)d_in[6];
  float* out = (float*)d_out;

  const size_t shmem = (size_t)(NWAVES * 8192 + 16 * D_ + NWAVES * 4 * D_ + 64) * sizeof(float);
  dim3 grid(B_ * KVH_);
  dim3 block(256);
  paged_attn_wmma_kernel<<<grid, block, shmem, stream>>>(q, kn, vn, kc, vc, bt, sl, out);
}